// weight_quantize_fn_47940424958076
// MI455X (gfx1250) — compile-verified
//
#include <hip/hip_runtime.h>

typedef __attribute__((ext_vector_type(2))) float v2f;
typedef __attribute__((ext_vector_type(8))) float v8f;

#define T1 256
#define EPT1 64
#define CHUNK1 (T1 * EPT1)   // 16384 elements per pass-1 block

#define T3 256
#define EPT3 16
#define CHUNK3 (T3 * EPT3)   // 4096 elements per pass-3 block

// ---------------------------------------------------------------------------
// Pass 1: per-block partial sum / sum-of-squares using WMMA row-sum trick.
// B = ones(4x16) => D[i][j] = rowsum(A)_i ; sum(all D entries) = 16 * sum(A).
// This identity is independent of the A/B/D VGPR striping, so it is robust.
// ---------------------------------------------------------------------------
__global__ __launch_bounds__(T1) void wq_pass1(const float* __restrict__ w,
                                               float* __restrict__ psum,
                                               float* __restrict__ psq,
                                               int n, int n_main) {
  const int tid = threadIdx.x;
  const int bid = blockIdx.x;

  v8f cs0 = {}, cs1 = {}, cq0 = {}, cq1 = {};
  const v2f ones = {1.0f, 1.0f};

  if ((bid + 1) * CHUNK1 <= n_main) {   // block-uniform branch: EXEC stays full
    const int base = bid * CHUNK1;
#pragma unroll
    for (int it = 0; it < 8; ++it) {
      const int off = base + it * (T1 * 8) + tid * 8;
      const float4 a = *(const float4*)(w + off);
      const float4 b = *(const float4*)(w + off + 4);
      v2f a0 = {a.x, a.y};
      v2f a1 = {a.z, a.w};
      v2f b0 = {b.x, b.y};
      v2f b1 = {b.z, b.w};
      cs0 = __builtin_amdgcn_wmma_f32_16x16x4_f32(false, a0, false, ones, (short)0, cs0, false, false);
      cs1 = __builtin_amdgcn_wmma_f32_16x16x4_f32(false, a1, false, ones, (short)0, cs1, false, false);
      cs0 = __builtin_amdgcn_wmma_f32_16x16x4_f32(false, b0, false, ones, (short)0, cs0, false, false);
      cs1 = __builtin_amdgcn_wmma_f32_16x16x4_f32(false, b1, false, ones, (short)0, cs1, false, false);
      cq0 = __builtin_amdgcn_wmma_f32_16x16x4_f32(false, a0 * a0, false, ones, (short)0, cq0, false, false);
      cq1 = __builtin_amdgcn_wmma_f32_16x16x4_f32(false, a1 * a1, false, ones, (short)0, cq1, false, false);
      cq0 = __builtin_amdgcn_wmma_f32_16x16x4_f32(false, b0 * b0, false, ones, (short)0, cq0, false, false);
      cq1 = __builtin_amdgcn_wmma_f32_16x16x4_f32(false, b1 * b1, false, ones, (short)0, cq1, false, false);
    }
  }

  float ps = 0.0f, pq = 0.0f;
#pragma unroll
  for (int k = 0; k < 8; ++k) {
    ps += cs0[k] + cs1[k];
    pq += cq0[k] + cq1[k];
  }
  ps *= 0.0625f;   // each row-sum is replicated over 16 columns of D
  pq *= 0.0625f;

  // tail (elements not covered by the WMMA-tiled main region) — block 0 only
  if (bid == 0) {
    for (int i = n_main + tid; i < n; i += T1) {
      const float x = w[i];
      ps += x;
      pq += x * x;
    }
  }

  // wave32 cross-lane reduction
#pragma unroll
  for (int o = 16; o > 0; o >>= 1) {
    ps += __shfl_xor(ps, o, 32);
    pq += __shfl_xor(pq, o, 32);
  }

  __shared__ float s_s[T1 / 32];
  __shared__ float s_q[T1 / 32];
  const int wave = tid >> 5;
  if ((tid & 31) == 0) { s_s[wave] = ps; s_q[wave] = pq; }
  __syncthreads();
  if (tid == 0) {
    float S = 0.0f, Q = 0.0f;
#pragma unroll
    for (int i = 0; i < T1 / 32; ++i) { S += s_s[i]; Q += s_q[i]; }
    psum[bid] = S;
    psq[bid] = Q;
  }
}

// ---------------------------------------------------------------------------
// Pass 2: reduce per-block partials -> mean, 1/(std*alpha)   (single block)
// ---------------------------------------------------------------------------
__global__ __launch_bounds__(256) void wq_pass2(const float* __restrict__ psum,
                                                const float* __restrict__ psq,
                                                const float* __restrict__ alphap,
                                                float* __restrict__ stats,
                                                int nblocks, int n) {
  __shared__ float ss[256];
  __shared__ float sq[256];
  const int tid = threadIdx.x;
  float s = 0.0f, q = 0.0f;
  for (int i = tid; i < nblocks; i += 256) { s += psum[i]; q += psq[i]; }
  ss[tid] = s;
  sq[tid] = q;
  __syncthreads();
  for (int st = 128; st > 0; st >>= 1) {
    if (tid < st) { ss[tid] += ss[tid + st]; sq[tid] += sq[tid + st]; }
    __syncthreads();
  }
  if (tid == 0) {
    const float S = ss[0];
    const float Q = sq[0];
    const float fn = (float)n;
    const float mean = S / fn;
    const float var = (Q - S * S / fn) / (fn - 1.0f);   // unbiased (ddof=1)
    const float stdv = sqrtf(var);
    const float alpha = alphap[0];
    stats[0] = mean;
    stats[1] = 1.0f / (stdv * alpha);
  }
}

// ---------------------------------------------------------------------------
// Pass 3: elementwise APoT quantization.
// Sorted 16-level codebook => nearest-with-first-min-tiebreak == descending
// select-cascade over the 15 midpoints (ax <= mid[i] picks the lower level).
// ---------------------------------------------------------------------------
__device__ __forceinline__ float wq_one(float x, float mean, float inv, float alpha) {
  constexpr float L[16] = {
      0.0f,
      0.015625f / 0.75f, 0.03125f  / 0.75f, 0.046875f / 0.75f,
      0.0625f   / 0.75f, 0.09375f  / 0.75f, 0.125f    / 0.75f,
      0.140625f / 0.75f, 0.1875f   / 0.75f, 0.25f     / 0.75f,
      0.28125f  / 0.75f, 0.375f    / 0.75f, 0.5f      / 0.75f,
      0.515625f / 0.75f, 0.5625f   / 0.75f, 1.0f};
  constexpr float M[15] = {
      (L[0]  + L[1])  * 0.5f, (L[1]  + L[2])  * 0.5f, (L[2]  + L[3])  * 0.5f,
      (L[3]  + L[4])  * 0.5f, (L[4]  + L[5])  * 0.5f, (L[5]  + L[6])  * 0.5f,
      (L[6]  + L[7])  * 0.5f, (L[7]  + L[8])  * 0.5f, (L[8]  + L[9])  * 0.5f,
      (L[9]  + L[10]) * 0.5f, (L[10] + L[11]) * 0.5f, (L[11] + L[12]) * 0.5f,
      (L[12] + L[13]) * 0.5f, (L[13] + L[14]) * 0.5f, (L[14] + L[15]) * 0.5f};

  float t = (x - mean) * inv;                  // ((w-mean)/std)/alpha
  t = fminf(1.0f, fmaxf(-1.0f, t));            // clip
  const float ax = fabsf(t);
  float q = L[15];
#pragma unroll
  for (int i = 14; i >= 0; --i) q = (ax <= M[i]) ? L[i] : q;
  return copysignf(q, t) * alpha;              // q==0 when t==0 (L[0]==0)
}

__global__ __launch_bounds__(T3) void wq_pass3(const float* __restrict__ w,
                                               const float* __restrict__ stats,
                                               const float* __restrict__ alphap,
                                               float* __restrict__ out, int n) {
  const float mean = stats[0];
  const float inv = stats[1];
  const float alpha = alphap[0];
  const int base = blockIdx.x * CHUNK3;
#pragma unroll
  for (int it = 0; it < EPT3 / 4; ++it) {
    const int off = base + it * (T3 * 4) + (int)threadIdx.x * 4;
    if (off + 4 <= n) {
      const float4 x = *(const float4*)(w + off);
      float4 r;
      r.x = wq_one(x.x, mean, inv, alpha);
      r.y = wq_one(x.y, mean, inv, alpha);
      r.z = wq_one(x.z, mean, inv, alpha);
      r.w = wq_one(x.w, mean, inv, alpha);
      *(float4*)(out + off) = r;
    } else {
      for (int i = off; i < n && i < off + 4; ++i)
        out[i] = wq_one(w[i], mean, inv, alpha);
    }
  }
}

// ---------------------------------------------------------------------------
extern "C" void kernel_launch(void* const* d_in, const int* in_sizes, int n_in,
                              void* d_out, int out_size, void* d_ws, size_t ws_size,
                              hipStream_t stream) {
  const float* w = (const float*)d_in[0];
  const float* alphap = (const float*)d_in[1];
  float* out = (float*)d_out;
  const int n = in_sizes[0];

  int blocks1 = n / CHUNK1;
  if (blocks1 < 1) blocks1 = 1;
  if (blocks1 > 2048) blocks1 = 2048;
  // make sure partials + stats fit in the workspace
  if (ws_size < (size_t)(2 * blocks1 + 2) * sizeof(float)) {
    int fit = (int)(ws_size / sizeof(float)) / 2 - 1;
    blocks1 = fit < 1 ? 1 : fit;
  }
  int n_main = blocks1 * CHUNK1;
  if (n_main > n) n_main = 0;   // tiny inputs: pure scalar tail path

  float* psum = (float*)d_ws;
  float* psq = psum + blocks1;
  float* stats = psq + blocks1;

  wq_pass1<<<blocks1, T1, 0, stream>>>(w, psum, psq, n, n_main);
  wq_pass2<<<1, 256, 0, stream>>>(psum, psq, alphap, stats, blocks1, n);
  const int blocks3 = (n + CHUNK3 - 1) / CHUNK3;
  wq_pass3<<<blocks3, T3, 0, stream>>>(w, stats, alphap, out, n);
}